// HorizontalROICropper_84104049590479
// MI455X (gfx1250) — compile-verified
//
#include <hip/hip_runtime.h>
#include <hip/hip_bf16.h>

#define H_IN   200
#define W_IN   320
#define C_IN   64
#define HOUT   48
#define WOUT   320
#define CH_STRIDE (H_IN * W_IN)   // 64000 floats between channels

// ---- gfx1250 feature probes (compile-time) ---------------------------------
#if defined(__AMDGCN__) && __has_builtin(__builtin_amdgcn_global_load_async_to_lds_b32)
  #define USE_ASYNC 1
#else
  #define USE_ASYNC 0
#endif

#if defined(__AMDGCN__)
  #if __has_builtin(__builtin_amdgcn_s_wait_asynccnt)
    #define WAIT_ASYNC(n) __builtin_amdgcn_s_wait_asynccnt(n)
  #else
    #define WAIT_ASYNC(n) asm volatile("s_wait_asynccnt %0" :: "i"(n) : "memory")
  #endif
#else
  #define WAIT_ASYNC(n) do {} while (0)
#endif

#if USE_ASYNC
__device__ __forceinline__ void async_copy_b32(const float* g, float* l) {
  // copies 4 bytes from the global address g to the LDS address l, tracked by ASYNCcnt (gfx1250 async-to-LDS path).
  // Builtin expects typed int* operands (AS1 global, AS3 local).
  __builtin_amdgcn_global_load_async_to_lds_b32(
      (__attribute__((address_space(1))) int*)g,
      (__attribute__((address_space(3))) int*)l,
      0 /*imm offset*/, 0 /*cpol*/);
}
#endif

// out[box][c][i][j] = x[img[box]][c][y0 + i*h/48][x0 + j*w/320]
__global__ __launch_bounds__(128)
void HorizontalROICropper_kernel(const float* __restrict__ x,
                                 const int*   __restrict__ bboxes,
                                 const int*   __restrict__ box_img,
                                 float*       __restrict__ out)
{
  const int box  = blockIdx.x;     // 0..255
  const int i    = blockIdx.y;     // 0..47
  const int wv   = threadIdx.x >> 5;   // wave id 0..3 (wave32)
  const int lane = threadIdx.x & 31;

  // Box parameters (uniform per block)
  const int b0 = bboxes[box * 4 + 0];
  const int b1 = bboxes[box * 4 + 1];
  const int b2 = bboxes[box * 4 + 2];
  const int b3 = bboxes[box * 4 + 3];
  const int img = box_img[box];

  const int x0 = min(max(b0, 0), W_IN - 1);
  const int y0 = min(max(b1, 0), H_IN - 1);
  const int x1 = max(min(max(b2, 0), W_IN - 1), x0);
  const int y1 = max(min(max(b3, 0), H_IN - 1), y0);
  const int w  = x1 - x0 + 1;                       // 1..320
  const int h  = y1 - y0 + 1;                       // 1..200
  const int row = y0 + (int)((unsigned)(i * h) / (unsigned)HOUT);

  // channel-0 source span start: x[img, 0, row, x0]
  const float* src0 = x + (size_t)img * (C_IN * (size_t)CH_STRIDE)
                        + (size_t)row * W_IN + x0;
  // output row base for c=0: out[box, 0, i, 0]
  float* outb = out + (size_t)box * (C_IN * (size_t)HOUT * WOUT)
                    + (size_t)i * WOUT;

  // Per-lane gather offsets within the span: col - x0 = j*w/320, j = lane+32u
  int coff[10];
  #pragma unroll
  for (int u = 0; u < 10; ++u) {
    const int jj = lane + 32 * u;
    coff[u] = (int)((unsigned)(jj * w) / (unsigned)WOUT);   // < w <= 320
  }

  const int cb = wv * 16;   // each wave owns 16 of the 64 channels

#if USE_ASYNC
  // Per-wave private double-buffered LDS slices: no barriers needed,
  // ASYNCcnt is tracked per wave.
  __shared__ float lds[4][2][WOUT];

  // Stage channel c's span into lds[wv][buf] (10 async b32 issues/wave;
  // lanes past the span re-load element w-1 so every issue is fully active,
  // keeping ASYNCcnt accounting constant at 10 per channel).
  #define STAGE(cc, buf)                                               \
    do {                                                               \
      const float* _s = src0 + (size_t)(cc) * CH_STRIDE;               \
      _Pragma("unroll")                                                \
      for (int _u = 0; _u < 10; ++_u) {                                \
        const int _t  = lane + 32 * _u;                                \
        const int _tc = (_t < w) ? _t : (w - 1);                       \
        async_copy_b32(_s + _tc, &lds[wv][buf][_t]);                   \
      }                                                                \
    } while (0)

  STAGE(cb, 0);
  for (int k = 0; k < 16; ++k) {
    const int c = cb + k;
    STAGE((k + 1 < 16) ? (c + 1) : c, (k + 1) & 1);  // prefetch next (dummy on last)
    WAIT_ASYNC(10);                                  // <=10 outstanding -> buf[k&1] ready
    float* oc = outb + (size_t)c * (HOUT * WOUT);
    const float* lb = lds[wv][k & 1];
    #pragma unroll
    for (int u = 0; u < 10; ++u) {
      // ds_load_b32 gather (bank-broadcast on duplicated columns),
      // nontemporal store: write-once output must not evict x from L2.
      __builtin_nontemporal_store(lb[coff[u]], oc + lane + 32 * u);
    }
  }
  #undef STAGE
#else
  // Fallback: direct cached gather (span lives in L0/L2; 192MB L2 holds all of x),
  // with explicit global_prefetch of the next channel's span.
  for (int k = 0; k < 16; ++k) {
    const int c = cb + k;
    const float* s = src0 + (size_t)c * CH_STRIDE;
    if (k + 1 < 16) {
      int p = lane * 10;
      if (p > w - 1) p = w - 1;                      // stay inside the valid span
      __builtin_prefetch(s + CH_STRIDE + p, 0, 1);   // global_prefetch_b8
    }
    float* oc = outb + (size_t)c * (HOUT * WOUT);
    #pragma unroll
    for (int u = 0; u < 10; ++u) {
      __builtin_nontemporal_store(s[coff[u]], oc + lane + 32 * u);
    }
  }
#endif
}

extern "C" void kernel_launch(void* const* d_in, const int* in_sizes, int n_in,
                              void* d_out, int out_size, void* d_ws, size_t ws_size,
                              hipStream_t stream) {
  const float* x       = (const float*)d_in[0];   // (8, 64, 200, 320) f32
  const int*   bboxes  = (const int*)d_in[1];     // (256, 4) i32
  const int*   box_img = (const int*)d_in[2];     // (256,) i32
  float*       out     = (float*)d_out;           // (256, 64, 48, 320) f32

  const int nboxes = in_sizes[2];                 // 256
  dim3 grid(nboxes, HOUT);                        // one block per (box, out_row)
  HorizontalROICropper_kernel<<<grid, 128, 0, stream>>>(x, bboxes, box_img, out);
}